// SimCLREEGAugmentor_84258668413046
// MI455X (gfx1250) — compile-verified
//
#include <hip/hip_runtime.h>
#include <stdint.h>

// SimCLR EEG augmentor for MI455X (gfx1250).
// x: [128, 64, 4096] f32.  out: [2, 128, 64, 4096] f32 (view1 ++ view2).
// One block per row-pair (r, r+4096); rows staged into LDS via the Tensor
// Data Mover (async tensor load, TENSORcnt), std computed in-block (roll-
// invariant), Threefry2x32 reproduces the JAX PRNG counter structure.

#define SLEN     4096
#define HALFROWS 4096          // rows 0..4095 pair with rows 4096..8191
#define NVIEW    33554432u     // 128*64*4096 elements per view
#define HALFN    16777216u     // NVIEW/2 : threefry halves-pairing offset

typedef uint32_t u32x4 __attribute__((ext_vector_type(4)));
typedef int      i32x4 __attribute__((ext_vector_type(4)));
typedef int      i32x8 __attribute__((ext_vector_type(8)));

__device__ __forceinline__ uint32_t rotl(uint32_t x, int r) {
  return (x << r) | (x >> (32 - r));
}

// Threefry-2x32, 20 rounds (JAX default PRNG core).
__device__ __forceinline__ void tf2x32(uint32_t k0, uint32_t k1,
                                       uint32_t c0, uint32_t c1,
                                       uint32_t& o0, uint32_t& o1) {
  const uint32_t k2 = k0 ^ k1 ^ 0x1BD11BDAu;
  uint32_t x0 = c0 + k0, x1 = c1 + k1;
#define TFR(r) { x0 += x1; x1 = rotl(x1, r); x1 ^= x0; }
  TFR(13) TFR(15) TFR(26) TFR(6)   x0 += k1; x1 += k2 + 1u;
  TFR(17) TFR(29) TFR(16) TFR(24)  x0 += k2; x1 += k0 + 2u;
  TFR(13) TFR(15) TFR(26) TFR(6)   x0 += k0; x1 += k1 + 3u;
  TFR(17) TFR(29) TFR(16) TFR(24)  x0 += k1; x1 += k2 + 4u;
  TFR(13) TFR(15) TFR(26) TFR(6)   x0 += k2; x1 += k0 + 5u;
#undef TFR
  o0 = x0; o1 = x1;
}

// JAX uniform[0,1): (bits>>9)|0x3f800000 bitcast, minus 1.
__device__ __forceinline__ float u01(uint32_t bits) {
  return __uint_as_float((bits >> 9) | 0x3f800000u) - 1.0f;
}

// ---- Tensor Data Mover: DMA one contiguous row of 4096 f32 into LDS. ----
// Descriptor layout per CDNA5 ISA ch.8 (D# groups). 6-arg builtin
// (amdgpu-toolchain clang-23): (u32x4, i32x8, i32x4, i32x4, i32x8, i32 cpol).
__device__ __forceinline__ void tdm_load_row_f32(const float* gsrc,
                                                 uint32_t lds_byte_addr) {
  const uint64_t ga = (uint64_t)(uintptr_t)gsrc;
  u32x4 g0;
  g0[0] = 1u;                                      // count=1, user mode
  g0[1] = lds_byte_addr;                           // lds_addr [63:32]
  g0[2] = (uint32_t)ga;                            // global_addr [95:64]
  g0[3] = (uint32_t)((ga >> 32) & 0x01FFFFFFu)     // global_addr [120:96]
        | (2u << 30);                              // type = 2 ("image")
  i32x8 g1;
  g1[0] = (int)(2u << 16);                         // data_size = 4 bytes
  g1[1] = (int)((uint32_t)SLEN << 16);             // tensor_dim0 = 4096
  g1[2] = (int)(1u << 16);                         // tensor_dim1 = 1
  g1[3] = (int)((uint32_t)SLEN << 16);             // tile_dim0 = 4096
  g1[4] = (int)1u;                                 // tile_dim1 = 1
  g1[5] = (int)(uint32_t)SLEN;                     // tensor_dim0_stride = 4096
  g1[6] = 0;
  g1[7] = 0;
  i32x4 z4; z4[0] = 0; z4[1] = 0; z4[2] = 0; z4[3] = 0;  // groups 2/3 unused
  i32x8 z8; z8[0] = 0; z8[1] = 0; z8[2] = 0; z8[3] = 0;
  z8[4] = 0; z8[5] = 0; z8[6] = 0; z8[7] = 0;            // extra slot unused
  __builtin_amdgcn_tensor_load_to_lds(g0, g1, z4, z4, z8, 0);
}

__global__ __launch_bounds__(256)
void simclr_eeg_aug_kernel(const float* __restrict__ x,
                           float* __restrict__ out) {
  __shared__ float ldsA[SLEN];   // row r       = (b, c),      b in [0,64)
  __shared__ float ldsB[SLEN];   // row r+4096  = (b+64, c)
  __shared__ float4 wpart[8];    // per-wave reduction partials

  const int r = blockIdx.x;      // 0..4095
  const int b = r >> 6;          // sample index of row A (row B: b+64)
  const int t = threadIdx.x;     // 256 threads = 8 waves

  // ---- stage both rows into LDS via TDM (issued by wave 0 only) ----
  if (t < 32) {
    tdm_load_row_f32(x + (size_t)r * SLEN,
                     (uint32_t)(uintptr_t)&ldsA[0]);
    tdm_load_row_f32(x + (size_t)(r + HALFROWS) * SLEN,
                     (uint32_t)(uintptr_t)&ldsB[0]);
    __builtin_amdgcn_s_wait_tensorcnt(0);
  }
  __syncthreads();

  // ---- per-row std (ddof=1); roll-invariant so computed pre-roll ----
  float sA = 0.f, qA = 0.f, sB = 0.f, qB = 0.f;
  for (int i = t; i < SLEN; i += 256) {
    float a = ldsA[i], c = ldsB[i];
    sA += a; qA = fmaf(a, a, qA);
    sB += c; qB = fmaf(c, c, qB);
  }
  for (int off = 16; off > 0; off >>= 1) {
    sA += __shfl_down(sA, off);
    qA += __shfl_down(qA, off);
    sB += __shfl_down(sB, off);
    qB += __shfl_down(qB, off);
  }
  if ((t & 31) == 0) wpart[t >> 5] = make_float4(sA, qA, sB, qB);
  __syncthreads();
  float ts = 0.f, tq = 0.f, us = 0.f, uq = 0.f;
#pragma unroll
  for (int w = 0; w < 8; ++w) {
    float4 p = wpart[w];
    ts += p.x; tq += p.y; us += p.z; uq += p.w;
  }
  const float inv_n   = 1.0f / (float)SLEN;
  const float inv_nm1 = 1.0f / (float)(SLEN - 1);
  const float stdA = sqrtf(fmaxf((tq - ts * ts * inv_n) * inv_nm1, 0.f));
  const float stdB = sqrtf(fmaxf((uq - us * us * inv_n) * inv_nm1, 0.f));

  // ---- two views ----
  for (int v = 0; v < 2; ++v) {
    // view key = fold_in(key(42), v+1) = threefry((0,42), (0,v+1))
    uint32_t vk0, vk1;
    tf2x32(0u, 42u, 0u, (uint32_t)(v + 1), vk0, vk1);
    // split(view_key, 4): bits = threefry(vk, iota(8)) halves-paired
    uint32_t w0, w1, w2, w3, w4, w5, w6, w7;
    tf2x32(vk0, vk1, 0u, 4u, w0, w4);
    tf2x32(vk0, vk1, 1u, 5u, w1, w5);
    tf2x32(vk0, vk1, 2u, 6u, w2, w6);
    tf2x32(vk0, vk1, 3u, 7u, w3, w7);
    // subkeys: k_shift=(w0,w1) k_noise=(w2,w3) k_mask=(w4,w5) k_scale=(w6,w7)

    // shifts: randint((128,), -50, 51); word pair (b, b+64)
    uint32_t sbA, sbB;
    tf2x32(w0, w1, (uint32_t)b, (uint32_t)(b + 64), sbA, sbB);
    const int shA = -50 + (int)(sbA % 101u);
    const int shB = -50 + (int)(sbB % 101u);

    // channel dropout mask: uniform((128,64)) > 0.1; word pair (r, r+4096)
    uint32_t mbA, mbB;
    tf2x32(w4, w5, (uint32_t)r, (uint32_t)(r + HALFROWS), mbA, mbB);
    const float maskA = (u01(mbA) > 0.1f) ? 1.f : 0.f;
    const float maskB = (u01(mbB) > 0.1f) ? 1.f : 0.f;

    // amplitude scale: uniform((128,), 0.8, 1.2); word pair (b, b+64)
    uint32_t cbA, cbB;
    tf2x32(w6, w7, (uint32_t)b, (uint32_t)(b + 64), cbA, cbB);
    const float gA = maskA * fmaf(u01(cbA), 0.4f, 0.8f);
    const float gB = maskB * fmaf(u01(cbB), 0.4f, 0.8f);

    const float nsA = 0.1f * stdA;
    const float nsB = 0.1f * stdB;

    float* __restrict__ oA = out + (size_t)v * NVIEW + (size_t)r * SLEN;
    float* __restrict__ oB = oA + (size_t)HALFROWS * SLEN;

    for (int iter = 0; iter < 4; ++iter) {
      const int s0 = iter * 1024 + t * 4;
      float va[4], vb[4];
#pragma unroll
      for (int e = 0; e < 4; ++e) {
        const int s = s0 + e;
        // noise word j pairs with j + N/2 -> covers rows r and r+4096
        const uint32_t j = (uint32_t)r * (uint32_t)SLEN + (uint32_t)s;
        uint32_t nA, nB;
        tf2x32(w2, w3, j, j + HALFN, nA, nB);
        float uA = fmaxf(fmaf(u01(nA), 2.0f, -1.0f), -0.99999994f);
        float uB = fmaxf(fmaf(u01(nB), 2.0f, -1.0f), -0.99999994f);
        const float zA = 1.41421356f * erfinvf(uA);   // sqrt(2)*erfinv
        const float zB = 1.41421356f * erfinvf(uB);
        // circular roll: out[s] = x[(s - shift) mod S]
        const float xa = ldsA[(uint32_t)(s - shA + SLEN) & (SLEN - 1)];
        const float xb = ldsB[(uint32_t)(s - shB + SLEN) & (SLEN - 1)];
        va[e] = (xa + zA * nsA) * gA;
        vb[e] = (xb + zB * nsB) * gB;
      }
      *(float4*)(oA + s0) = make_float4(va[0], va[1], va[2], va[3]);
      *(float4*)(oB + s0) = make_float4(vb[0], vb[1], vb[2], vb[3]);
    }
  }
}

extern "C" void kernel_launch(void* const* d_in, const int* in_sizes, int n_in,
                              void* d_out, int out_size, void* d_ws, size_t ws_size,
                              hipStream_t stream) {
  (void)in_sizes; (void)n_in; (void)out_size; (void)d_ws; (void)ws_size;
  const float* x = (const float*)d_in[0];
  float* out = (float*)d_out;
  simclr_eeg_aug_kernel<<<dim3(HALFROWS), dim3(256), 0, stream>>>(x, out);
}